// TokenBasedTTT_34187939676313
// MI455X (gfx1250) — compile-verified
//
#include <hip/hip_runtime.h>
#include <hip/hip_bf16.h>

// ---------------------------------------------------------------------------
// TokenBasedTTT for MI455X (gfx1250): fp32 WMMA pipeline.
//   K1: x = hidden @ W_in + b_in            (wmma f32 16x16x4), fused row
//       sum / sum-of-squares reduction (shfl_xor + global atomic f32)
//   K2: c[t] = exclusive cumsum of 0.01*mean_b ||x_bt||^2
//   K3: pre = x @ fast_weights + fb + c*rowsum ; LayerNorm ; @ W_out + b_out
// LDS staging uses gfx1250 async memory->LDS copies when available.
// ---------------------------------------------------------------------------

typedef float v2f __attribute__((ext_vector_type(2)));
typedef float v8f __attribute__((ext_vector_type(8)));
typedef int   v4i __attribute__((ext_vector_type(4)));

#define WMMA_F32(a, b, c) \
  __builtin_amdgcn_wmma_f32_16x16x4_f32(false, (a), false, (b), (short)0, (c), false, false)

constexpr int kB  = 16;
constexpr int kT  = 4096;
constexpr int kDW = 1024;
constexpr int kD  = 768;
constexpr int kV  = 32;
constexpr int kM  = kB * kT;          // 65536 rows total
constexpr float kLR  = 0.01f;
constexpr float kEPS = 1e-5f;

constexpr int LDA = kDW + 4;          // padded LDS stride for A strip (banks)
constexpr int LDP = kD + 4;           // padded LDS stride for x/pre strips

// ---- gfx1250 async global->LDS staging ------------------------------------
#if defined(__AMDGCN__) && __has_builtin(__builtin_amdgcn_global_load_async_to_lds_b128)
#define TTT_ASYNC_LDS 1
#else
#define TTT_ASYNC_LDS 0
#endif

__device__ __forceinline__ void stage16(const void* gsrc, void* ldst) {
#if TTT_ASYNC_LDS
  __builtin_amdgcn_global_load_async_to_lds_b128(
      (__attribute__((address_space(1))) v4i*)gsrc,
      (__attribute__((address_space(3))) v4i*)ldst,
      /*offset=*/0, /*cpol=*/0);
#else
  *(float4*)ldst = *(const float4*)gsrc;
#endif
}

__device__ __forceinline__ void stage_wait() {
#if TTT_ASYNC_LDS
#if __has_builtin(__builtin_amdgcn_s_wait_asynccnt)
  __builtin_amdgcn_s_wait_asynccnt(0);
#else
  asm volatile("s_wait_asynccnt 0x0" ::: "memory");
#endif
#endif
}

// ---------------------------------------------------------------------------
__global__ __launch_bounds__(256) void ttt_zero_f32(float* __restrict__ p, int n) {
  int i = blockIdx.x * blockDim.x + threadIdx.x;
  if (i < n) p[i] = 0.0f;
}

// ---------------------------------------------------------------------------
// Kernel 1: GEMM1 (M=65536, N=768, K=1024) + per-row sum / sumsq reduction.
// One workgroup (8 waves) per 16-row strip. A strip staged in LDS.
// Each wave owns 6 N-tiles, processed as 3 pairs sharing one A load per
// K-step (2 independent WMMAs per A fetch).
__global__ __launch_bounds__(256)
void ttt_gemm1(const float* __restrict__ hidden, const float* __restrict__ W_in,
               const float* __restrict__ b_in, float* __restrict__ x,
               float* __restrict__ rowsum, float* __restrict__ s_raw) {
  extern __shared__ float ldsA[];                 // 16 x LDA floats
  const int strip = blockIdx.x;                   // 0..4095
  const int tid   = threadIdx.x;
  const int lane  = tid & 31;
  const int wave  = tid >> 5;
  const int mrow  = lane & 15;                    // A row within tile
  const int khalf = lane >> 4;                    // 0: K{0,1}, 1: K{2,3}

  // Stage A strip: 16 rows x 1024 cols (async b128, padded LDS stride).
  {
    const float4* src = (const float4*)(hidden + (size_t)strip * 16 * kDW);
    for (int i = tid; i < 16 * (kDW / 4); i += 256) {
      int r  = i >> 8;                            // 256 float4 per row
      int c4 = i & 255;
      stage16(&src[(size_t)r * (kDW / 4) + c4], &ldsA[r * LDA + c4 * 4]);
    }
  }
  stage_wait();
  __syncthreads();

  float rs[8], ss[8];
#pragma unroll
  for (int r = 0; r < 8; ++r) { rs[r] = 0.0f; ss[r] = 0.0f; }

  // 48 N-tiles / 8 waves = 6 per wave = 3 pairs.
  for (int np = 0; np < 3; ++np) {
    const int nt0  = wave * 6 + np * 2;
    const int col0 = nt0 * 16 + (lane & 15);
    const int col1 = col0 + 16;
    const float bias0 = b_in[col0];
    const float bias1 = b_in[col1];
    v8f acc0, acc1;
#pragma unroll
    for (int r = 0; r < 8; ++r) { acc0[r] = bias0; acc1[r] = bias1; }

    const float* __restrict__ Bp0 = W_in + col0;  // stride kD per K row
    const float* __restrict__ Bp1 = W_in + col1;
#pragma unroll 4
    for (int k0 = 0; k0 < kDW; k0 += 4) {
      const int k = k0 + khalf * 2;
      v2f a, b0, b1;
      a[0]  = ldsA[mrow * LDA + k];
      a[1]  = ldsA[mrow * LDA + k + 1];
      b0[0] = Bp0[(size_t)k * kD];
      b0[1] = Bp0[(size_t)(k + 1) * kD];
      b1[0] = Bp1[(size_t)k * kD];
      b1[1] = Bp1[(size_t)(k + 1) * kD];
      acc0 = WMMA_F32(a, b0, acc0);
      acc1 = WMMA_F32(a, b1, acc1);
    }

#pragma unroll
    for (int r = 0; r < 8; ++r) {
      const int row = strip * 16 + r + 8 * khalf;
      const float v0 = acc0[r], v1 = acc1[r];
      x[(size_t)row * kD + col0] = v0;
      x[(size_t)row * kD + col1] = v1;
      rs[r] += v0 + v1;
      ss[r] += v0 * v0 + v1 * v1;
    }
  }

  // Reduce across the 16 lanes of each half (C/D layout: row = r + 8*khalf).
#pragma unroll
  for (int r = 0; r < 8; ++r) {
    float a = rs[r], q = ss[r];
#pragma unroll
    for (int off = 1; off < 16; off <<= 1) {
      a += __shfl_xor(a, off, 32);
      q += __shfl_xor(q, off, 32);
    }
    if ((lane & 15) == 0) {
      const int row = strip * 16 + r + 8 * khalf;
      atomicAdd(&rowsum[row], a);
      atomicAdd(&s_raw[row & (kT - 1)], q);       // t = row % T (T = 4096)
    }
  }
}

// ---------------------------------------------------------------------------
// Kernel 2: c[t] = exclusive cumsum of (LR/B) * s_raw[t], t in [0,4096).
__global__ __launch_bounds__(256)
void ttt_scan(const float* __restrict__ s_raw, float* __restrict__ c) {
  __shared__ float part[256];
  const int tid = threadIdx.x;
  const float scale = kLR / (float)kB;
  float local[16];
  float sum = 0.0f;
#pragma unroll
  for (int i = 0; i < 16; ++i) {
    float v = s_raw[tid * 16 + i] * scale;
    local[i] = v;
    sum += v;
  }
  part[tid] = sum;
  __syncthreads();
  for (int off = 1; off < 256; off <<= 1) {
    float add = (tid >= off) ? part[tid - off] : 0.0f;
    __syncthreads();
    part[tid] += add;
    __syncthreads();
  }
  float run = (tid == 0) ? 0.0f : part[tid - 1];  // exclusive chunk prefix
#pragma unroll
  for (int i = 0; i < 16; ++i) {
    c[tid * 16 + i] = run;
    run += local[i];
  }
}

// ---------------------------------------------------------------------------
// Kernel 3: pre = x @ fast_weights + fb + c*rowsum ; LayerNorm ; @ W_out.
__global__ __launch_bounds__(256)
void ttt_fused2(const float* __restrict__ x, const float* __restrict__ fw,
                const float* __restrict__ fb, const float* __restrict__ gamma,
                const float* __restrict__ beta, const float* __restrict__ W_out,
                const float* __restrict__ b_out, const float* __restrict__ rowsum,
                const float* __restrict__ c, float* __restrict__ out) {
  extern __shared__ float lds[];
  float* xs    = lds;                 // 16 * LDP
  float* ps    = xs + 16 * LDP;       // 16 * LDP
  float* rowc  = ps + 16 * LDP;       // 16
  float* rowS  = rowc + 16;           // 16
  float* mu_s  = rowS + 16;           // 16
  float* rsd_s = mu_s + 16;           // 16

  const int strip = blockIdx.x;
  const int tid   = threadIdx.x;
  const int lane  = tid & 31;
  const int wave  = tid >> 5;
  const int mrow  = lane & 15;
  const int khalf = lane >> 4;

  // Stage x strip (16 x 768) into padded LDS (async b128).
  for (int r = 0; r < 16; ++r) {
    const float4* src = (const float4*)(x + (size_t)(strip * 16 + r) * kD);
    for (int j = tid; j < kD / 4; j += 256)
      stage16(&src[j], &xs[r * LDP + j * 4]);
  }
  if (tid < 16) {
    const int row = strip * 16 + tid;
    rowS[tid] = rowsum[row];
    rowc[tid] = c[row & (kT - 1)];
  }
  stage_wait();
  __syncthreads();

  // GEMM2: pre[16,768], 6 N-tiles per wave as 3 pairs, K = 768.
  for (int np = 0; np < 3; ++np) {
    const int nt0  = wave * 6 + np * 2;
    const int col0 = nt0 * 16 + (lane & 15);
    const int col1 = col0 + 16;
    const float bias0 = fb[col0];
    const float bias1 = fb[col1];
    v8f acc0, acc1;
#pragma unroll
    for (int r = 0; r < 8; ++r) {
      const int lr = r + 8 * khalf;
      const float t = rowc[lr] * rowS[lr];
      acc0[r] = bias0 + t;
      acc1[r] = bias1 + t;
    }
    const float* __restrict__ Bp0 = fw + col0;
    const float* __restrict__ Bp1 = fw + col1;
#pragma unroll 4
    for (int k0 = 0; k0 < kD; k0 += 4) {
      const int k = k0 + khalf * 2;
      v2f a, b0, b1;
      a[0]  = xs[mrow * LDP + k];
      a[1]  = xs[mrow * LDP + k + 1];
      b0[0] = Bp0[(size_t)k * kD];
      b0[1] = Bp0[(size_t)(k + 1) * kD];
      b1[0] = Bp1[(size_t)k * kD];
      b1[1] = Bp1[(size_t)(k + 1) * kD];
      acc0 = WMMA_F32(a, b0, acc0);
      acc1 = WMMA_F32(a, b1, acc1);
    }
#pragma unroll
    for (int r = 0; r < 8; ++r) {
      ps[(r + 8 * khalf) * LDP + col0] = acc0[r];
      ps[(r + 8 * khalf) * LDP + col1] = acc1[r];
    }
  }
  __syncthreads();

  // LayerNorm: 16 threads per row; two-pass (mean, then centered variance)
  // because the c*rowsum term is a large per-row constant.
  {
    const int row = tid >> 4, j0 = tid & 15;
    float s = 0.0f;
    for (int j = j0; j < kD; j += 16) s += ps[row * LDP + j];
#pragma unroll
    for (int off = 1; off < 16; off <<= 1) s += __shfl_xor(s, off, 32);
    const float mu = s * (1.0f / (float)kD);
    float v = 0.0f;
    for (int j = j0; j < kD; j += 16) {
      float d = ps[row * LDP + j] - mu;
      v += d * d;
    }
#pragma unroll
    for (int off = 1; off < 16; off <<= 1) v += __shfl_xor(v, off, 32);
    if (j0 == 0) {
      mu_s[row]  = mu;
      rsd_s[row] = rsqrtf(v * (1.0f / (float)kD) + kEPS);
    }
  }
  __syncthreads();

  // Normalize in place with gamma/beta.
  {
    const int row = tid >> 4, j0 = tid & 15;
    const float mu = mu_s[row], rstd = rsd_s[row];
    for (int j = j0; j < kD; j += 16)
      ps[row * LDP + j] = (ps[row * LDP + j] - mu) * rstd * gamma[j] + beta[j];
  }
  __syncthreads();

  // GEMM3: logits[16,32] = out_strip @ W_out + b_out. Waves 0 and 1 only.
  if (wave < 2) {
    const int nt  = wave;
    const int col = nt * 16 + (lane & 15);
    v8f acc;
    const float bias = b_out[col];
#pragma unroll
    for (int r = 0; r < 8; ++r) acc[r] = bias;
    const float* __restrict__ Bp = W_out + col;
#pragma unroll 4
    for (int k0 = 0; k0 < kD; k0 += 4) {
      const int k = k0 + khalf * 2;
      v2f a, b;
      a[0] = ps[mrow * LDP + k];
      a[1] = ps[mrow * LDP + k + 1];
      b[0] = Bp[(size_t)k * kV];
      b[1] = Bp[(size_t)(k + 1) * kV];
      acc = WMMA_F32(a, b, acc);
    }
#pragma unroll
    for (int r = 0; r < 8; ++r) {
      const int row = strip * 16 + r + 8 * khalf;
      out[(size_t)row * kV + col] = acc[r];
    }
  }
}

// ---------------------------------------------------------------------------
extern "C" void kernel_launch(void* const* d_in, const int* in_sizes, int n_in,
                              void* d_out, int out_size, void* d_ws, size_t ws_size,
                              hipStream_t stream) {
  const float* hidden = (const float*)d_in[0];
  const float* W_in   = (const float*)d_in[1];
  const float* b_in   = (const float*)d_in[2];
  const float* fw     = (const float*)d_in[3];
  const float* fb     = (const float*)d_in[4];
  const float* gamma  = (const float*)d_in[5];
  const float* beta   = (const float*)d_in[6];
  const float* W_out  = (const float*)d_in[7];
  const float* b_out  = (const float*)d_in[8];
  float* out = (float*)d_out;

  char* ws = (char*)d_ws;
  float* x      = (float*)ws;                                // 65536*768 f32
  float* rowsum = (float*)(ws + (size_t)kM * kD * sizeof(float));  // 65536
  float* s_raw  = rowsum + kM;                               // 4096
  float* c      = s_raw + kT;                                // 4096

  // Zero accumulators (rowsum + s_raw are contiguous).
  {
    const int n = kM + kT;
    ttt_zero_f32<<<(n + 255) / 256, 256, 0, stream>>>(rowsum, n);
  }

  ttt_gemm1<<<dim3(kM / 16), dim3(256), 16 * LDA * sizeof(float), stream>>>(
      hidden, W_in, b_in, x, rowsum, s_raw);

  ttt_scan<<<1, 256, 0, stream>>>(s_raw, c);

  ttt_fused2<<<dim3(kM / 16), dim3(256),
               (2 * 16 * LDP + 64) * sizeof(float), stream>>>(
      x, fw, fb, gamma, beta, W_out, b_out, rowsum, c, out);
}